// GCN_21174188769405
// MI455X (gfx1250) — compile-verified
//
#include <hip/hip_runtime.h>
#include <math.h>

typedef __attribute__((ext_vector_type(2))) float v2f;
typedef __attribute__((ext_vector_type(8))) float v8f;

#define BN_EPS  1e-5f
#define GEN_EPS 1e-7f

// ---------------------------------------------------------------------------
// Degree (in-degree over dst, +1 self loop), then dinv = rsqrt(deg)
// ---------------------------------------------------------------------------
__global__ void deg_init_kernel(float* __restrict__ deg, int n) {
  int i = blockIdx.x * blockDim.x + threadIdx.x;
  if (i < n) deg[i] = 1.0f;  // self loop
}

__global__ void deg_count_kernel(const long long* __restrict__ ei,
                                 float* __restrict__ deg, int E) {
  int e = blockIdx.x * blockDim.x + threadIdx.x;
  if (e < E) atomicAdd(&deg[(int)ei[(size_t)E + e]], 1.0f);
}

__global__ void deg_rsqrt_kernel(float* __restrict__ deg, int n) {
  int i = blockIdx.x * blockDim.x + threadIdx.x;
  if (i < n) deg[i] = rsqrtf(deg[i]);
}

// ---------------------------------------------------------------------------
// FP32 GEMM via V_WMMA_F32_16X16X4_F32.
// C[M,NC] = A[M,K] @ B[K,NC] (+ bias). M%16==0, K%4==0, NC%64==0 or NC==64.
// Block = 128 threads = 4 waves; each wave owns one 16x16 tile; block covers
// 16 rows x 64 cols. grid = (M/16, NC/64).
// A-operand ISA layout (32-bit 16x4): lanes 0-15 -> K=k0,k0+1 of row M=lane;
// lanes 16-31 -> K=k0+2,k0+3.  B mirrored.  C/D: v[r] = row r (+8 for hi half).
// ---------------------------------------------------------------------------
__global__ void gemm_f32_wmma(const float* __restrict__ A,
                              const float* __restrict__ B,
                              const float* __restrict__ bias,
                              float* __restrict__ C,
                              int M, int K, int NC) {
  const int wave = threadIdx.x >> 5;
  const int lane = threadIdx.x & 31;
  const int l15  = lane & 15;
  const int hi   = lane >> 4;              // 0 or 1
  const int row0 = blockIdx.x * 16;
  const int col0 = blockIdx.y * 64 + wave * 16;

  const float* Arow = A + (size_t)(row0 + l15) * K;
  v8f acc = {};
  for (int k0 = 0; k0 < K; k0 += 4) {
    const int ka = k0 + 2 * hi;
    v2f a, b;
    a.x = Arow[ka];
    a.y = Arow[ka + 1];
    b.x = B[(size_t)ka * NC + col0 + l15];
    b.y = B[(size_t)(ka + 1) * NC + col0 + l15];
    acc = __builtin_amdgcn_wmma_f32_16x16x4_f32(false, a, false, b,
                                                (short)0, acc, false, false);
  }

  const int colw  = col0 + l15;
  const float bv  = bias ? bias[colw] : 0.0f;
  const int rbase = row0 + hi * 8;
#pragma unroll
  for (int r = 0; r < 8; ++r)
    C[(size_t)(rbase + r) * NC + colw] = acc[r] + bv;
}

// ---------------------------------------------------------------------------
// GCNConv aggregation: out = b + dinv[i]^2 * h[i] (self loop) then
// atomic scatter of dinv[s]*dinv[d]*h[s] over edges.  C fixed at 128.
// ---------------------------------------------------------------------------
__global__ void gcn_self_init_kernel(const float* __restrict__ h,
                                     const float* __restrict__ dinv,
                                     const float* __restrict__ bias,
                                     float* __restrict__ out, int n) {
  size_t idx = (size_t)blockIdx.x * blockDim.x + threadIdx.x;
  size_t total = (size_t)n * 128;
  if (idx >= total) return;
  int i = (int)(idx >> 7);
  int c = (int)(idx & 127);
  float di = dinv[i];
  out[idx] = bias[c] + di * di * h[idx];
}

__global__ void gcn_scatter_kernel(const float* __restrict__ h,
                                   const float* __restrict__ dinv,
                                   const long long* __restrict__ ei,
                                   float* __restrict__ out, int E) {
  size_t tid = (size_t)blockIdx.x * blockDim.x + threadIdx.x;
  int e = (int)(tid >> 5);
  if (e >= E) return;
  int q = ((int)tid & 31) * 4;                  // 4 channels per lane
  int s = (int)ei[e];
  int d = (int)ei[(size_t)E + e];
  float w = dinv[s] * dinv[d];
  const float4 hv = *(const float4*)(h + (size_t)s * 128 + q);
  float* o = out + (size_t)d * 128 + q;
  atomicAdd(o + 0, w * hv.x);
  atomicAdd(o + 1, w * hv.y);
  atomicAdd(o + 2, w * hv.z);
  atomicAdd(o + 3, w * hv.w);
}

// ---------------------------------------------------------------------------
// BatchNorm (training stats, biased variance) + ReLU
// ---------------------------------------------------------------------------
__global__ void bn_stats_kernel(const float* __restrict__ h,
                                float* __restrict__ stats, int n, int C) {
  int c = threadIdx.x;                       // blockDim.x == C
  float s = 0.0f, sq = 0.0f;
  for (int i = blockIdx.x; i < n; i += gridDim.x) {
    float v = h[(size_t)i * C + c];
    s += v;
    sq += v * v;
  }
  atomicAdd(&stats[c], s);
  atomicAdd(&stats[C + c], sq);
}

__global__ void bn_finalize_kernel(const float* __restrict__ stats,
                                   const float* __restrict__ gamma,
                                   const float* __restrict__ beta,
                                   float* __restrict__ ss, int n, int C) {
  int c = blockIdx.x * blockDim.x + threadIdx.x;
  if (c >= C) return;
  float inv_n = 1.0f / (float)n;
  float mu  = stats[c] * inv_n;
  float var = stats[C + c] * inv_n - mu * mu;
  float sc  = gamma[c] * rsqrtf(var + BN_EPS);
  ss[c]     = sc;
  ss[C + c] = beta[c] - mu * sc;
}

__global__ void bn_apply_relu_kernel(const float* __restrict__ in,
                                     const float* __restrict__ ss,
                                     float* __restrict__ out,
                                     size_t total, int C) {
  size_t idx = (size_t)blockIdx.x * blockDim.x + threadIdx.x;
  if (idx >= total) return;
  int c = (int)(idx % (size_t)C);
  float v = in[idx] * ss[c] + ss[C + c];
  out[idx] = v > 0.0f ? v : 0.0f;
}

// ---------------------------------------------------------------------------
// GENConv softmax aggregation.  x is post-ReLU so msg = x[src] + eps > 0,
// letting us do segment-max with uint atomicMax (bit order == float order
// for non-negative floats; m buffer is zero-initialized).
// ---------------------------------------------------------------------------
__global__ void gen_max_kernel(const float* __restrict__ x,
                               const long long* __restrict__ ei,
                               unsigned int* __restrict__ m, int E) {
  size_t tid = (size_t)blockIdx.x * blockDim.x + threadIdx.x;
  int e = (int)(tid >> 5);
  if (e >= E) return;
  int q = ((int)tid & 31) * 4;
  int s = (int)ei[e];
  int d = (int)ei[(size_t)E + e];
  const float4 v = *(const float4*)(x + (size_t)s * 128 + q);
  unsigned int* mo = m + (size_t)d * 128 + q;
  atomicMax(mo + 0, __float_as_uint(v.x + GEN_EPS));
  atomicMax(mo + 1, __float_as_uint(v.y + GEN_EPS));
  atomicMax(mo + 2, __float_as_uint(v.z + GEN_EPS));
  atomicMax(mo + 3, __float_as_uint(v.w + GEN_EPS));
}

__global__ void gen_sum_kernel(const float* __restrict__ x,
                               const long long* __restrict__ ei,
                               const float* __restrict__ m,
                               float* __restrict__ denom,
                               float* __restrict__ num, int E) {
  size_t tid = (size_t)blockIdx.x * blockDim.x + threadIdx.x;
  int e = (int)(tid >> 5);
  if (e >= E) return;
  int q = ((int)tid & 31) * 4;
  int s = (int)ei[e];
  int d = (int)ei[(size_t)E + e];
  const float4 v  = *(const float4*)(x + (size_t)s * 128 + q);
  const float4 mv = *(const float4*)(m + (size_t)d * 128 + q);
  float* de = denom + (size_t)d * 128 + q;
  float* nu = num   + (size_t)d * 128 + q;
  float msg, ex;
  msg = v.x + GEN_EPS; ex = expf(msg - mv.x);
  atomicAdd(de + 0, ex); atomicAdd(nu + 0, ex * msg);
  msg = v.y + GEN_EPS; ex = expf(msg - mv.y);
  atomicAdd(de + 1, ex); atomicAdd(nu + 1, ex * msg);
  msg = v.z + GEN_EPS; ex = expf(msg - mv.z);
  atomicAdd(de + 2, ex); atomicAdd(nu + 2, ex * msg);
  msg = v.w + GEN_EPS; ex = expf(msg - mv.w);
  atomicAdd(de + 3, ex); atomicAdd(nu + 3, ex * msg);
}

__global__ void gen_combine_kernel(const float* __restrict__ num,
                                   const float* __restrict__ den,
                                   const float* __restrict__ x,
                                   float* __restrict__ out, size_t total) {
  size_t idx = (size_t)blockIdx.x * blockDim.x + threadIdx.x;
  if (idx >= total) return;
  out[idx] = num[idx] / fmaxf(den[idx], 1e-16f) + x[idx];
}

// ---------------------------------------------------------------------------
// Row-wise log-softmax over 64 classes; one wave32 per row, 2 cols per lane.
// ---------------------------------------------------------------------------
__global__ void logsoftmax64_kernel(float* __restrict__ out, int n) {
  int row  = blockIdx.x * (blockDim.x >> 5) + (threadIdx.x >> 5);
  int lane = threadIdx.x & 31;
  if (row >= n) return;
  float* r = out + (size_t)row * 64;
  float a = r[lane];
  float b = r[lane + 32];
  float mx = fmaxf(a, b);
#pragma unroll
  for (int off = 16; off > 0; off >>= 1) mx = fmaxf(mx, __shfl_xor(mx, off));
  float s = expf(a - mx) + expf(b - mx);
#pragma unroll
  for (int off = 16; off > 0; off >>= 1) s += __shfl_xor(s, off);
  float L = mx + logf(s);
  r[lane]      = a - L;
  r[lane + 32] = b - L;
}

// ---------------------------------------------------------------------------
// Host-side orchestration
// ---------------------------------------------------------------------------
extern "C" void kernel_launch(void* const* d_in, const int* in_sizes, int n_in,
                              void* d_out, int out_size, void* d_ws, size_t ws_size,
                              hipStream_t stream) {
  const float* x     = (const float*)d_in[0];
  const float* W0    = (const float*)d_in[1];
  const float* b0    = (const float*)d_in[2];
  const float* g0    = (const float*)d_in[3];
  const float* beta0 = (const float*)d_in[4];
  const float* W1    = (const float*)d_in[5];
  const float* b1    = (const float*)d_in[6];
  const float* g1    = (const float*)d_in[7];
  const float* beta1 = (const float*)d_in[8];
  const float* Wg1   = (const float*)d_in[9];
  const float* bg1   = (const float*)d_in[10];
  const float* gg    = (const float*)d_in[11];
  const float* betag = (const float*)d_in[12];
  const float* Wg2   = (const float*)d_in[13];
  const float* bg2   = (const float*)d_in[14];
  const long long* ei = (const long long*)d_in[15];

  const int N = in_sizes[0] / 128;
  const int E = in_sizes[15] / 2;

  float* ws     = (float*)d_ws;
  float* dinv   = ws;                       // [N]
  float* stats  = dinv + N;                 // [1024]: raw [0..511], ss [512..1023]
  float* ssbuf  = stats + 512;
  float* bufA   = stats + 1024;             // [N,128]
  float* bufB   = bufA + (size_t)N * 128;   // [N,128]
  float* bufC   = bufB + (size_t)N * 128;   // [N,128]
  float* bufD   = bufC + (size_t)N * 128;   // [N,128]
  float* buf256 = bufD + (size_t)N * 128;   // [N,256]
  float* outp   = (float*)d_out;            // [N,64]

  const int TB = 256;
  const int nBlkN    = (N + TB - 1) / TB;
  const int nBlkE    = (E + TB - 1) / TB;
  const int nBlkE32  = (int)(((size_t)E * 32 + TB - 1) / TB);
  const int nBlkNC   = (int)(((size_t)N * 128 + TB - 1) / TB);
  const int nBlkNC2  = (int)(((size_t)N * 256 + TB - 1) / TB);
  const size_t featB = (size_t)N * 128 * sizeof(float);

  // --- degrees ---
  deg_init_kernel<<<nBlkN, TB, 0, stream>>>(dinv, N);
  deg_count_kernel<<<nBlkE, TB, 0, stream>>>(ei, dinv, E);
  deg_rsqrt_kernel<<<nBlkN, TB, 0, stream>>>(dinv, N);

  // --- layer 0: GCNConv -> BN -> ReLU ---
  gemm_f32_wmma<<<dim3(N / 16, 2), 128, 0, stream>>>(x, W0, nullptr, bufA, N, 128, 128);
  gcn_self_init_kernel<<<nBlkNC, TB, 0, stream>>>(bufA, dinv, b0, bufB, N);
  gcn_scatter_kernel<<<nBlkE32, TB, 0, stream>>>(bufA, dinv, ei, bufB, E);
  hipMemsetAsync(stats, 0, 512 * sizeof(float), stream);
  bn_stats_kernel<<<512, 128, 0, stream>>>(bufB, stats, N, 128);
  bn_finalize_kernel<<<1, 128, 0, stream>>>(stats, g0, beta0, ssbuf, N, 128);
  bn_apply_relu_kernel<<<nBlkNC, TB, 0, stream>>>(bufB, ssbuf, bufA, (size_t)N * 128, 128);

  // --- layer 1: GCNConv -> BN -> ReLU ---
  gemm_f32_wmma<<<dim3(N / 16, 2), 128, 0, stream>>>(bufA, W1, nullptr, bufB, N, 128, 128);
  gcn_self_init_kernel<<<nBlkNC, TB, 0, stream>>>(bufB, dinv, b1, bufA, N);
  gcn_scatter_kernel<<<nBlkE32, TB, 0, stream>>>(bufB, dinv, ei, bufA, E);
  hipMemsetAsync(stats, 0, 512 * sizeof(float), stream);
  bn_stats_kernel<<<512, 128, 0, stream>>>(bufA, stats, N, 128);
  bn_finalize_kernel<<<1, 128, 0, stream>>>(stats, g1, beta1, ssbuf, N, 128);
  bn_apply_relu_kernel<<<nBlkNC, TB, 0, stream>>>(bufA, ssbuf, bufB, (size_t)N * 128, 128);
  // bufB = x fed to GENConv (post-ReLU, all >= 0)

  // --- GENConv softmax aggregation ---
  hipMemsetAsync(bufA, 0, featB, stream);        // m
  hipMemsetAsync(bufC, 0, 2 * featB, stream);    // denom + num (contiguous)
  gen_max_kernel<<<nBlkE32, TB, 0, stream>>>(bufB, ei, (unsigned int*)bufA, E);
  gen_sum_kernel<<<nBlkE32, TB, 0, stream>>>(bufB, ei, bufA, bufC, bufD, E);
  gen_combine_kernel<<<nBlkNC, TB, 0, stream>>>(bufD, bufC, bufB, bufA, (size_t)N * 128);

  // --- GEN MLP: [128] -> [256] (BN+ReLU) -> [64] ---
  gemm_f32_wmma<<<dim3(N / 16, 4), 128, 0, stream>>>(bufA, Wg1, bg1, buf256, N, 128, 256);
  hipMemsetAsync(stats, 0, 512 * sizeof(float), stream);
  bn_stats_kernel<<<512, 256, 0, stream>>>(buf256, stats, N, 256);
  bn_finalize_kernel<<<1, 256, 0, stream>>>(stats, gg, betag, ssbuf, N, 256);
  bn_apply_relu_kernel<<<nBlkNC2, TB, 0, stream>>>(buf256, ssbuf, buf256, (size_t)N * 256, 256);
  gemm_f32_wmma<<<dim3(N / 16, 1), 128, 0, stream>>>(buf256, Wg2, bg2, outp, N, 256, 64);

  // --- log-softmax over 64 classes ---
  logsoftmax64_kernel<<<(N + 7) / 8, 256, 0, stream>>>(outp, N);
}